// GAT_89859305766919
// MI455X (gfx1250) — compile-verified
//
#include <hip/hip_runtime.h>
#include <hip/hip_bf16.h>

typedef __attribute__((ext_vector_type(2))) float v2f;
typedef __attribute__((ext_vector_type(8))) float v8f;

#define NNODES 100000
#define NEDGES 1600000

// ---------------------------------------------------------------------------
// WMMA f32 GEMM:  D[N,C] = A[N,K] * B[K,C]
// One wave computes a 16x16 output tile using V_WMMA_F32_16X16X4_F32.
// Block = (C/16) waves, grid = N/16 row tiles. N is an exact multiple of 16
// (100000 = 6250*16) so EXEC stays all-ones (WMMA requirement).
//
// f32 A-fragment layout (ISA 7.12.2, 16x4 MxK, 2 VGPRs):
//   lane in [0,15]:  m=lane,    v0=K0, v1=K1
//   lane in [16,31]: m=lane-16, v0=K2, v1=K3
// B-fragment (4x16 KxN) mirrors with lane = column n.
// D layout: VGPR i -> row (i + 8*(lane>>4)), col (lane&15).
// ---------------------------------------------------------------------------
template <int K, int C>
__global__ __launch_bounds__(32 * (C / 16)) void gemm_wmma_kernel(
    const float* __restrict__ A, const float* __restrict__ B,
    float* __restrict__ D) {
  const int lane = threadIdx.x & 31;
  const int wave = threadIdx.x >> 5;        // column tile within block
  const int row0 = blockIdx.x * 16;
  const int col0 = wave * 16;
  const int half = lane >> 4;               // 0 or 1
  const int l    = lane & 15;               // m for A frag, n for B frag

  const float* arow = A + (long)(row0 + l) * K + 2 * half;
  const float* bcol = B + col0 + l;

  v8f acc = {};
#pragma unroll 4
  for (int k = 0; k < K; k += 4) {
    v2f a, b;
    a.x = arow[k + 0];
    a.y = arow[k + 1];
    b.x = bcol[(k + 2 * half + 0) * C];
    b.y = bcol[(k + 2 * half + 1) * C];
    acc = __builtin_amdgcn_wmma_f32_16x16x4_f32(
        /*neg_a=*/false, a, /*neg_b=*/false, b,
        /*c_mod=*/(short)0, acc, /*reuse_a=*/false, /*reuse_b=*/false);
  }

  float* drow = D + (long)(row0 + 8 * half) * C + col0 + l;
#pragma unroll
  for (int i = 0; i < 8; ++i) drow[i * C] = acc[i];
}

// ---------------------------------------------------------------------------
// fill 32-bit pattern
// ---------------------------------------------------------------------------
__global__ void fill_u32_kernel(unsigned* __restrict__ p, unsigned v, int n) {
  int i = blockIdx.x * blockDim.x + threadIdx.x;
  if (i < n) p[i] = v;
}

// ---------------------------------------------------------------------------
// attention logits: el[n,h] = sum_d feat[n,h,d]*al[h,d]; er likewise
// ---------------------------------------------------------------------------
template <int H, int D>
__global__ void logits_kernel(const float* __restrict__ feat,
                              const float* __restrict__ al,
                              const float* __restrict__ ar,
                              float* __restrict__ el, float* __restrict__ er,
                              int NH) {
  int i = blockIdx.x * blockDim.x + threadIdx.x;
  if (i >= NH) return;
  int n = i / H, h = i % H;
  const float* f = feat + ((long)n * H + h) * D;
  const float* a = al + h * D;
  const float* r = ar + h * D;
  float sl = 0.f, sr = 0.f;
#pragma unroll
  for (int d = 0; d < D; ++d) {
    sl += f[d] * a[d];
    sr += f[d] * r[d];
  }
  el[i] = sl;
  er[i] = sr;
}

__device__ __forceinline__ unsigned float_ord_encode(float f) {
  unsigned u = __float_as_uint(f);
  return (u & 0x80000000u) ? ~u : (u | 0x80000000u);
}
__device__ __forceinline__ float float_ord_decode(unsigned u) {
  return (u & 0x80000000u) ? __uint_as_float(u ^ 0x80000000u)
                           : __uint_as_float(~u);
}
#define ENC_NEG_INF 0x007FFFFFu  // encode(-inf)

// ---------------------------------------------------------------------------
// pass 1: per-dst per-head segment max of leaky_relu(el[src]+er[dst])
// ---------------------------------------------------------------------------
template <int H>
__global__ void edge_max_kernel(const int* __restrict__ src,
                                const int* __restrict__ dst,
                                const float* __restrict__ el,
                                const float* __restrict__ er,
                                unsigned* __restrict__ mu, int EH) {
  int i = blockIdx.x * blockDim.x + threadIdx.x;
  if (i >= EH) return;
  int e = i / H, h = i % H;
  int s = src[e], d = dst[e];
  float v = el[s * H + h] + er[d * H + h];
  v = v > 0.f ? v : 0.2f * v;
  atomicMax(&mu[d * H + h], float_ord_encode(v));
}

// ---------------------------------------------------------------------------
// pass 2: w = exp(e - m[dst]); denom[dst] += w; agg[dst,:] += w*feat[src,:]
// ---------------------------------------------------------------------------
template <int H, int D>
__global__ void edge_agg_kernel(const int* __restrict__ src,
                                const int* __restrict__ dst,
                                const float* __restrict__ el,
                                const float* __restrict__ er,
                                const unsigned* __restrict__ mu,
                                float* __restrict__ den,
                                const float* __restrict__ feat,
                                float* __restrict__ agg, int EH) {
  int i = blockIdx.x * blockDim.x + threadIdx.x;
  if (i >= EH) return;
  int e = i / H, h = i % H;
  int s = src[e], d = dst[e];
  float v = el[s * H + h] + er[d * H + h];
  v = v > 0.f ? v : 0.2f * v;
  float m = float_ord_decode(mu[d * H + h]);
  float w = expf(v - m);
  atomicAdd(&den[d * H + h], w);
  const float* fs = feat + ((long)s * H + h) * D;
  float* ad = agg + ((long)d * H + h) * D;
#pragma unroll
  for (int k = 0; k < D; ++k) atomicAdd(&ad[k], w * fs[k]);
}

// ---------------------------------------------------------------------------
// normalize + bias (+ optional ELU)
// ---------------------------------------------------------------------------
template <int H, int D, bool DO_ELU>
__global__ void finalize_kernel(const float* __restrict__ agg,
                                const float* __restrict__ den,
                                const float* __restrict__ bias,
                                float* __restrict__ out, int NC) {
  int i = blockIdx.x * blockDim.x + threadIdx.x;
  if (i >= NC) return;
  int n = i / (H * D);
  int c = i % (H * D);
  int h = c / D;
  float dn = den[n * H + h];
  float x = dn > 0.f ? agg[i] / dn : 0.f;
  x += bias[c];
  if (DO_ELU) x = x > 0.f ? x : expf(x) - 1.f;
  out[i] = x;
}

extern "C" void kernel_launch(void* const* d_in, const int* in_sizes, int n_in,
                              void* d_out, int out_size, void* d_ws,
                              size_t ws_size, hipStream_t stream) {
  const float* x   = (const float*)d_in[0];
  const int*   src = (const int*)d_in[1];
  const int*   dst = (const int*)d_in[2];
  const float* W1  = (const float*)d_in[3];
  const float* al1 = (const float*)d_in[4];
  const float* ar1 = (const float*)d_in[5];
  const float* b1  = (const float*)d_in[6];
  const float* W2  = (const float*)d_in[7];
  const float* al2 = (const float*)d_in[8];
  const float* ar2 = (const float*)d_in[9];
  const float* b2  = (const float*)d_in[10];
  float* out = (float*)d_out;

  const int N = NNODES, E = NEDGES;

  // workspace layout (floats), regions overlapped across phases:
  float* ws = (float*)d_ws;
  float*    feat1 = ws;                         // N*64  (reused as h1)
  float*    el1   = feat1 + (long)N * 64;       // N*8
  float*    er1   = el1 + (long)N * 8;          // N*8
  unsigned* m1u   = (unsigned*)(er1 + (long)N * 8);  // N*8
  float*    den1  = (float*)(m1u + (long)N * 8);     // N*8
  float*    agg1  = den1 + (long)N * 8;         // N*64
  // layer-2 aliases (valid only after layer-1 finalize):
  float*    h1    = feat1;                      // ELU output, N*64
  float*    feat2 = agg1;                       // N*32 (agg1 dead)
  unsigned* m2u   = (unsigned*)(agg1 + (long)N * 32);  // N
  float*    den2  = agg1 + (long)N * 32 + N;           // N
  float*    el2   = agg1 + (long)N * 32 + 2 * N;       // N
  float*    er2   = agg1 + (long)N * 32 + 3 * N;       // N
  float*    agg2  = el1;  // N*32 (el1/er1/m1u/den1 dead after finalize1)

  const int TB = 256;
  auto nb = [](long n, int b) { return (int)((n + b - 1) / b); };

  // ---------------- layer 1 ----------------
  fill_u32_kernel<<<nb((long)N * 8, TB), TB, 0, stream>>>(m1u, ENC_NEG_INF, N * 8);
  fill_u32_kernel<<<nb((long)N * 8, TB), TB, 0, stream>>>((unsigned*)den1, 0u, N * 8);
  fill_u32_kernel<<<nb((long)N * 64, TB), TB, 0, stream>>>((unsigned*)agg1, 0u, N * 64);

  gemm_wmma_kernel<256, 64><<<N / 16, 128, 0, stream>>>(x, W1, feat1);
  logits_kernel<8, 8><<<nb((long)N * 8, TB), TB, 0, stream>>>(feat1, al1, ar1, el1, er1, N * 8);
  edge_max_kernel<8><<<nb((long)E * 8, TB), TB, 0, stream>>>(src, dst, el1, er1, m1u, E * 8);
  edge_agg_kernel<8, 8><<<nb((long)E * 8, TB), TB, 0, stream>>>(src, dst, el1, er1, m1u, den1, feat1, agg1, E * 8);
  finalize_kernel<8, 8, true><<<nb((long)N * 64, TB), TB, 0, stream>>>(agg1, den1, b1, h1, N * 64);

  // ---------------- layer 2 ----------------
  // (fills must come after finalize1: agg2 aliases el1..den1, m2u/den2 alias agg1 tail)
  fill_u32_kernel<<<nb((long)N, TB), TB, 0, stream>>>(m2u, ENC_NEG_INF, N);
  fill_u32_kernel<<<nb((long)N, TB), TB, 0, stream>>>((unsigned*)den2, 0u, N);

  gemm_wmma_kernel<64, 32><<<N / 16, 64, 0, stream>>>(h1, W2, feat2);
  logits_kernel<1, 32><<<nb((long)N, TB), TB, 0, stream>>>(feat2, al2, ar2, el2, er2, N);

  fill_u32_kernel<<<nb((long)N * 32, TB), TB, 0, stream>>>((unsigned*)agg2, 0u, N * 32);

  edge_max_kernel<1><<<nb((long)E, TB), TB, 0, stream>>>(src, dst, el2, er2, m2u, E);
  edge_agg_kernel<1, 32><<<nb((long)E, TB), TB, 0, stream>>>(src, dst, el2, er2, m2u, den2, feat2, agg2, E);
  finalize_kernel<1, 32, false><<<nb((long)N * 32, TB), TB, 0, stream>>>(agg2, den2, b2, out, N * 32);
}